// PatchLoss_34651796144543
// MI455X (gfx1250) — compile-verified
//
#include <hip/hip_runtime.h>
#include <hip/hip_bf16.h>
#include <math.h>

// ---------------------------------------------------------------------------
// PatchLoss on MI455X (gfx1250, wave32)
//
// x, x_r: (1, 3, 32, 512, 512) f32.  Output: scalar f32.
// Memory-bound: ~200 MB read / 23.3 TB/s => ~8.6 us floor. Strategy:
//   - b128 (float4) streaming loads, 8192 blocks x 256 threads (1 block/patch)
//   - wave-level sum via V_WMMA_F32_16X16X4_F32 with B = ones (exact f32)
//   - tiny epilogue kernel for per-frame max / clamp / mean / log
// ---------------------------------------------------------------------------

typedef __attribute__((ext_vector_type(2))) float v2f;
typedef __attribute__((ext_vector_type(8))) float v8f;

// Sum a float across all 32 lanes of a wave using one f32 WMMA.
// A: lane L holds its value at (M = L%16, K = L<16 ? 0 : 2); K=1,3 are zero.
// B = ones(4x16)  =>  D[m][n] = v[m] + v[m+16] for every n.
// Lane layout of D (8 VGPRs): lanes 0-15 see rows 0-7 in c[0..7],
// lanes 16-31 see rows 8-15. Sum c[0..7] then xor-16 shuffle => full wave sum.
__device__ __forceinline__ float wave_sum_wmma(float v) {
    v2f a; a.x = v;    a.y = 0.0f;
    v2f b; b.x = 1.0f; b.y = 1.0f;
    v8f c = {};
    c = __builtin_amdgcn_wmma_f32_16x16x4_f32(
        /*neg_a=*/false, a, /*neg_b=*/false, b,
        /*c_mod=*/(short)0, c, /*reuse_a=*/false, /*reuse_b=*/false);
    float s = c[0] + c[1] + c[2] + c[3] + c[4] + c[5] + c[6] + c[7];
    s += __shfl_xor(s, 16, 32);
    return s;
}

// Tensor geometry (compile-time constants from the reference shapes)
#define T_STRIDE   262144       // 512*512
#define C_STRIDE   8388608      // 32*512*512
#define W_DIM      512
#define PATCH      32           // hc = wc = 32
#define NPH        16           // 512/32
#define PATCH_ELEMS 3072.0f     // 3 * 32 * 32

// One block per (t, ph, pw) patch: 32*16*16 = 8192 blocks, 256 threads.
// Each thread: one float4 per channel per tensor (6 loads), 3072 elems total.
__global__ void __launch_bounds__(256)
patch_sum_kernel(const float* __restrict__ x,
                 const float* __restrict__ xr,
                 float* __restrict__ P) {
    const int pid = blockIdx.x;            // t*256 + ph*16 + pw
    const int t   = pid >> 8;
    const int ph  = (pid >> 4) & 15;
    const int pw  = pid & 15;
    const int tid = threadIdx.x;           // 0..255
    const int r   = tid >> 3;              // row within patch, 0..31
    const int g   = tid & 7;               // float4 group within row, 0..7

    const int base = t * T_STRIDE + (ph * PATCH + r) * W_DIM + pw * PATCH + g * 4;

    // Issue all 6 b128 loads up front for memory-level parallelism.
    const float4 a0 = *(const float4*)(x  + base);
    const float4 b0 = *(const float4*)(xr + base);
    const float4 a1 = *(const float4*)(x  + base + C_STRIDE);
    const float4 b1 = *(const float4*)(xr + base + C_STRIDE);
    const float4 a2 = *(const float4*)(x  + base + 2 * C_STRIDE);
    const float4 b2 = *(const float4*)(xr + base + 2 * C_STRIDE);

    float acc = 0.0f;
    acc += fabsf(a0.x - b0.x) + fabsf(a0.y - b0.y) + fabsf(a0.z - b0.z) + fabsf(a0.w - b0.w);
    acc += fabsf(a1.x - b1.x) + fabsf(a1.y - b1.y) + fabsf(a1.z - b1.z) + fabsf(a1.w - b1.w);
    acc += fabsf(a2.x - b2.x) + fabsf(a2.y - b2.y) + fabsf(a2.z - b2.z) + fabsf(a2.w - b2.w);

    // Wave-level sum via WMMA (EXEC all-ones here: no divergence yet).
    const float ws = wave_sum_wmma(acc);

    __shared__ float lds[8];
    const int wave = tid >> 5;
    const int lane = tid & 31;
    if (lane == 0) lds[wave] = ws;
    __syncthreads();

    if (tid == 0) {
        float s = 0.0f;
#pragma unroll
        for (int w = 0; w < 8; ++w) s += lds[w];
        // L = |x - x_r| / 2, patch mean over 3072 elements
        P[pid] = s * (0.5f / PATCH_ELEMS);
    }
}

// Single block, 32 waves (1024 threads). Wave w = frame t = w:
// max over its 256 patch means (8 per lane + xor-shuffle tree),
// then wave 0 clamps at 0, averages over 32 frames, takes log.
__global__ void __launch_bounds__(1024)
patch_max_log_kernel(const float* __restrict__ P, float* __restrict__ out) {
    const int tid  = threadIdx.x;
    const int wave = tid >> 5;
    const int lane = tid & 31;

    __shared__ float tmax[32];

    const float* Pt = P + wave * 256;
    float m = -INFINITY;
#pragma unroll
    for (int i = 0; i < 8; ++i) m = fmaxf(m, Pt[lane + i * 32]);
#pragma unroll
    for (int off = 16; off > 0; off >>= 1) m = fmaxf(m, __shfl_xor(m, off, 32));
    if (lane == 0) tmax[wave] = m;
    __syncthreads();

    if (wave == 0) {
        float v = fmaxf(tmax[lane], 0.0f);   // reference clamps max_patch at 0
#pragma unroll
        for (int off = 16; off > 0; off >>= 1) v += __shfl_xor(v, off, 32);
        if (lane == 0) out[0] = logf(v * (1.0f / 32.0f));
    }
}

extern "C" void kernel_launch(void* const* d_in, const int* in_sizes, int n_in,
                              void* d_out, int out_size, void* d_ws, size_t ws_size,
                              hipStream_t stream) {
    const float* x  = (const float*)d_in[0];
    const float* xr = (const float*)d_in[1];
    float* P   = (float*)d_ws;    // 8192 floats of scratch (32 KB)
    float* out = (float*)d_out;

    patch_sum_kernel<<<8192, 256, 0, stream>>>(x, xr, P);
    patch_max_log_kernel<<<1, 1024, 0, stream>>>(P, out);
}